// TransformDomainInterpolator_10101763080845
// MI455X (gfx1250) — compile-verified
//
#include <hip/hip_runtime.h>

// ---------------------------------------------------------------------------
// TransformDomainInterpolator on MI455X (gfx1250, wave32, WMMA)
//
//   F[2m]   = x[m]                                    (exact pass-through)
//   F[2m+1] = (1/2048) * IDFT2048( s_k * e^{i*pi*k/2048} * DFT2048(x) )[m]
//   out[b,t,f] = (1-u_t)*F_sym2 + u_t*F_sym11,  u_t = clamp((t-2)/9, 0, 1)
//
// DFT2048 = radix-8 (VALU) * radix-16 (WMMA) * radix-16 (WMMA); the radix-16
// stages are 16x16 complex matmuls on V_WMMA_F32_16X16X4_F32. All twiddles
// come from one 2048-entry LDS table (built once per block), so the VALU mix
// is FMAs + table lookups instead of transcendental expansions.
// ---------------------------------------------------------------------------

typedef __attribute__((ext_vector_type(2))) float v2f;
typedef __attribute__((ext_vector_type(4))) float v4f;
typedef __attribute__((ext_vector_type(8))) float v8f;

__device__ inline v8f wmma4(v2f a, v2f b, v8f c) {
  // (neg_a, A, neg_b, B, c_mod, C, reuse_a, reuse_b)
  return __builtin_amdgcn_wmma_f32_16x16x4_f32(false, a, false, b, (short)0, c,
                                               false, false);
}

// Accumulate one 16x16 complex tile D = W * Z into (accRe, accIm).
//   W[M][K] = tw[(K * (128*M + aOff)) & 2047]   (conjugated when INV)
//   Z element (K,N) at src[srcBase + K*sK + N*sN]
template <bool INV>
__device__ inline void dft16_accum(const float2* __restrict__ src,
                                   const float2* __restrict__ tw,
                                   int srcBase, int sK, int sN, int aOff,
                                   v8f& accRe, v8f& accIm) {
  const int lane = threadIdx.x & 31;
  const int h = lane >> 4;   // half-wave
  const int nl = lane & 15;  // M for A, N for B/C/D
  const int aRow = 128 * nl + aOff;
#pragma unroll
  for (int j = 0; j < 4; ++j) {
    const int k0 = 4 * j + 2 * h;
    const int k1 = k0 + 1;
    const float2 t0 = tw[(k0 * aRow) & 2047];
    const float2 t1 = tw[(k1 * aRow) & 2047];
    v2f are, aim, aimn;
    are.x = t0.x;
    are.y = t1.x;
    aim.x = INV ? -t0.y : t0.y;
    aim.y = INV ? -t1.y : t1.y;
    aimn.x = -aim.x;
    aimn.y = -aim.y;
    const float2 b0 = src[srcBase + k0 * sK + nl * sN];
    const float2 b1 = src[srcBase + k1 * sK + nl * sN];
    v2f bre, bim;
    bre.x = b0.x; bre.y = b1.x;
    bim.x = b0.y; bim.y = b1.y;
    accRe = wmma4(are, bre, accRe);   // Re += Wre*Zre
    accRe = wmma4(aimn, bim, accRe);  // Re -= Wim*Zim
    accIm = wmma4(are, bim, accIm);   // Im += Wre*Zim
    accIm = wmma4(aim, bre, accIm);   // Im += Wim*Zre
  }
}

// Store a tile; optional level-1 twiddle tw[(N*(pT + 8*M)) & 2047] applied.
template <bool INV, bool POST>
__device__ inline void dft16_store(float2* __restrict__ dst,
                                   const float2* __restrict__ tw,
                                   int dstBase, int dM, int dN, int pT,
                                   v8f accRe, v8f accIm) {
  const int lane = threadIdx.x & 31;
  const int h = lane >> 4;
  const int nl = lane & 15;
#pragma unroll
  for (int r = 0; r < 8; ++r) {
    const int mrow = r + 8 * h;
    float re = accRe[r], im = accIm[r];
    if (POST) {
      const float2 tp = tw[(nl * (pT + 8 * mrow)) & 2047];
      const float c = tp.x;
      const float s = INV ? -tp.y : tp.y;
      const float r2 = re * c - im * s;
      im = re * s + im * c;
      re = r2;
    }
    dst[dstBase + mrow * dM + nl * dN] = make_float2(re, im);
  }
}

template <bool INV, bool POST>
__device__ inline void dft16_stage(const float2* __restrict__ src,
                                   float2* __restrict__ dst,
                                   const float2* __restrict__ tw, int srcBase,
                                   int sK, int sN, int dstBase, int dM, int dN,
                                   int aOff, int pT) {
  v8f accRe = {}, accIm = {};
  dft16_accum<INV>(src, tw, srcBase, sK, sN, aOff, accRe, accIm);
  dft16_store<INV, POST>(dst, tw, dstBase, dM, dN, pT, accRe, accIm);
}

// Radix-8 stage: 256 threads, one (n1,m1) group each.
// C[n1][m1][c2] = sum_m2 in[n1+16*m1+256*m2] * exp(sgn*2*pi*i*m2*c2/8)
// MODSPEC fuses the spectral reshape (sign-split * half-bin shift * 1/2048).
template <bool INV, bool MODSPEC>
__device__ inline void radix8_stage(const float2* __restrict__ src,
                                    float2* __restrict__ dst, int tid) {
  const float sgn = INV ? 1.f : -1.f;
  const float C8[8] = {1.f, 0.70710678f, 0.f, -0.70710678f,
                       -1.f, -0.70710678f, 0.f, 0.70710678f};
  const float S8[8] = {0.f, 0.70710678f, 1.f, 0.70710678f,
                       0.f, -0.70710678f, -1.f, -0.70710678f};
  const int base = (tid & 15) + 16 * (tid >> 4);
  float2 xv[8];
#pragma unroll
  for (int m2 = 0; m2 < 8; ++m2) {
    const int k = base + 256 * m2;
    float2 v = src[k];
    if (MODSPEC) {
      float ss, cc;
      sincospif((float)k * (1.0f / 2048.0f), &ss, &cc);
      const float sc = (k < 1024 ? 1.0f : -1.0f) * (1.0f / 2048.0f);
      const float c = cc * sc;
      const float s = ss * sc;
      v = make_float2(v.x * c - v.y * s, v.x * s + v.y * c);
    }
    xv[m2] = v;
  }
#pragma unroll
  for (int c2 = 0; c2 < 8; ++c2) {
    float re = 0.f, im = 0.f;
#pragma unroll
    for (int m2 = 0; m2 < 8; ++m2) {
      const int j = (m2 * c2) & 7;
      const float c = C8[j];
      const float s = sgn * S8[j];
      re += xv[m2].x * c - xv[m2].y * s;
      im += xv[m2].x * s + xv[m2].y * c;
    }
    dst[base + 256 * c2] = make_float2(re, im);
  }
}

// One block per (symbol s, batch b): 2x band-limited upsample -> F[4096]
__global__ __launch_bounds__(256) void tdi_transform_kernel(
    const float* __restrict__ hre, const float* __restrict__ him,
    float2* __restrict__ F) {
  __shared__ float2 xs[2048];  // pilots (= even outputs)
  __shared__ float2 ys[2048];  // twiddle table, then odd outputs
  __shared__ float2 w0[2048];
  __shared__ float2 w1[2048];
  float2* tw = ys;  // table aliases ys until the final (guarded) store

  const int tid = threadIdx.x;
  const int wave = tid >> 5;  // 8 waves -> 8 tiles per WMMA stage
  const int b = blockIdx.x >> 1;
  const int s = blockIdx.x & 1;

  for (int j = tid; j < 2048; j += 256) {
    float ss, cc;
    sincospif((float)j * (1.0f / 1024.0f), &ss, &cc);
    tw[j] = make_float2(cc, -ss);  // e^{-2*pi*i*j/2048}
  }
  const float* pr = hre + (size_t)b * 4096 + s * 2048;
  const float* pi = him + (size_t)b * 4096 + s * 2048;
  for (int m = tid; m < 2048; m += 256) xs[m] = make_float2(pr[m], pi[m]);
  __syncthreads();

  // ---- forward DFT2048
  radix8_stage<false, false>(xs, w0, tid);
  __syncthreads();
  dft16_stage<false, true>(w0, w1, tw, 256 * wave, 16, 1, 16 * wave, 128, 1,
                           16 * wave, wave);
  __syncthreads();
  dft16_stage<false, false>(w1, w0, tw, 256 * wave, 1, 16, 16 * wave, 128, 1,
                            0, 0);
  __syncthreads();

  // ---- inverse DFT2048 with fused spectral reshape on load
  radix8_stage<true, true>(w0, w1, tid);
  __syncthreads();
  dft16_stage<true, true>(w1, w0, tw, 256 * wave, 16, 1, 16 * wave, 128, 1,
                          16 * wave, wave);
  __syncthreads();
  // Final stage writes ys (aliases the table): read table, barrier, store.
  {
    v8f accRe = {}, accIm = {};
    dft16_accum<true>(w0, tw, 256 * wave, 1, 16, 0, accRe, accIm);
    __syncthreads();
    dft16_store<true, false>(ys, nullptr, 16 * wave, 128, 1, 0, accRe, accIm);
  }
  __syncthreads();

  // ---- interleave even(=pilots)/odd(=interpolated) as 16B vectors
  v4f* Fout = (v4f*)(F + (size_t)((s << 9) + b) * 4096);
  for (int m = tid; m < 2048; m += 256) {
    v4f v;
    v.x = xs[m].x; v.y = xs[m].y;
    v.z = ys[m].x; v.w = ys[m].y;
    Fout[m] = v;
  }
}

// Linear time interpolation across the 14 OFDM symbols; F stays L2-resident,
// the 234MB output is streamed with non-temporal stores.
__global__ __launch_bounds__(256) void tdi_timeinterp_kernel(
    const v4f* __restrict__ F4, v4f* __restrict__ out4) {
  const int gid = blockIdx.x * 256 + threadIdx.x;  // [0, 512*2048)
  const int b = gid >> 11;
  const int fp = gid & 2047;
  const v4f f2 = F4[(size_t)b * 2048 + fp];
  const v4f f11 = F4[(size_t)(512 + b) * 2048 + fp];
  const v4f d = f11 - f2;
  v4f* po = out4 + (size_t)b * (14 * 2048) + fp;
#pragma unroll
  for (int t = 0; t < 14; ++t) {
    const float u =
        (t < 2) ? 0.f : ((t >= 11) ? 1.f : (float)(t - 2) * (1.f / 9.f));
    const v4f v = f2 + d * u;
    __builtin_nontemporal_store(v, po + (size_t)t * 2048);
  }
}

extern "C" void kernel_launch(void* const* d_in, const int* in_sizes, int n_in,
                              void* d_out, int out_size, void* d_ws,
                              size_t ws_size, hipStream_t stream) {
  const float* hre = (const float*)d_in[0];
  const float* him = (const float*)d_in[1];
  float2* F = (float2*)d_ws;     // 2*512*4096 complex64 = 33.5 MB scratch
  tdi_transform_kernel<<<1024, 256, 0, stream>>>(hre, him, F);
  tdi_timeinterp_kernel<<<(512 * 2048) / 256, 256, 0, stream>>>(
      (const v4f*)d_ws, (v4f*)d_out);
}